// MultiPositionTransfer_40553081209240
// MI455X (gfx1250) — compile-verified
//
#include <hip/hip_runtime.h>
#include <math.h>

typedef float v2f __attribute__((ext_vector_type(2)));
typedef float v8f __attribute__((ext_vector_type(8)));

#define NTOK   65536      // T*B = 2048*32
#define NEXP   9          // multi_infer_num + 1
#define DEMB   128
#define MTILE  64         // tokens per block (4 m-subtiles of 16)
#define MAXTILES (NTOK / MTILE)   // 1024 (worst case: all tokens on one expert)

__device__ __forceinline__ float sigm(float x) {
    return 1.0f / (1.0f + __expf(-x));
}

// ---------------- routing kernels (expert bucketing in d_ws) ----------------

__global__ void mpt_init_kernel(int* counts, int* fill) {
    int t = threadIdx.x;
    if (t < NEXP) { counts[t] = 0; fill[t] = 0; }
}

__global__ void mpt_count_kernel(const int* __restrict__ pos, int* __restrict__ counts) {
    int tid = blockIdx.x * blockDim.x + threadIdx.x;
    if (tid < NTOK) {
        int p = pos[tid];
        int e = p < (NEXP - 1) ? p : (NEXP - 1);
        e = e < 0 ? 0 : e;
        atomicAdd(&counts[e], 1);
    }
}

__global__ void mpt_scan_kernel(const int* __restrict__ counts, int* __restrict__ offsets) {
    if (threadIdx.x == 0 && blockIdx.x == 0) {
        int run = 0;
        for (int e = 0; e < NEXP; ++e) { offsets[e] = run; run += counts[e]; }
    }
}

__global__ void mpt_scatter_kernel(const int* __restrict__ pos,
                                   const int* __restrict__ offsets,
                                   int* __restrict__ fill,
                                   int* __restrict__ tok) {
    int tid = blockIdx.x * blockDim.x + threadIdx.x;
    if (tid < NTOK) {
        int p = pos[tid];
        int e = p < (NEXP - 1) ? p : (NEXP - 1);
        e = e < 0 ? 0 : e;
        int slot = atomicAdd(&fill[e], 1);
        tok[offsets[e] + slot] = tid;
    }
}

// ---------------- grouped-GEMM + in-register gating layer kernel ----------------
// grid: (MAXTILES, NEXP); block: 256 threads = 8 wave32, 64-token tile.
// wave w -> f-quarter fq = w&3 (32 f values, 2 n-tiles)
//           m-pair     mp = w>>2 (2 m-subtiles of 16 tokens)
// Each wave computes ALL FOUR linears for its (m,f) slice, so the
// sigmoid/tanh gating happens entirely in registers (no LDS round-trip):
// for fixed (lane, acc-row r, nt) the four accumulators are the same (token,f).

__global__ __launch_bounds__(256) void mpt_layer_kernel(
    const float* __restrict__ xin, float* __restrict__ xout,
    const float* __restrict__ Ws, const float* __restrict__ bs,
    const int* __restrict__ counts, const int* __restrict__ offsets,
    const int* __restrict__ tok, int layer)
{
    const int e = blockIdx.y;
    const int seg_len = counts[e];
    const int tile0 = blockIdx.x * MTILE;
    if (tile0 >= seg_len) return;
    const int seg_start = offsets[e];

    __shared__ float xs[MTILE][DEMB + 4];   // +4 pad: conflict-free b64 A-frag reads
    __shared__ int   tokLDS[MTILE];

    const int tid  = threadIdx.x;
    const int w    = tid >> 5;          // wave id 0..7
    const int lane = tid & 31;

    if (tid < MTILE) {
        int r = tile0 + tid;
        tokLDS[tid] = (r < seg_len) ? tok[seg_start + r] : -1;
    }
    __syncthreads();

    // Gather A tile: 64 token rows x 128 f32. One wave per row per pass:
    // 32 lanes x float4 = full 512B row, coalesced.
    #pragma unroll
    for (int it = 0; it < MTILE / 8; ++it) {
        int row = w + it * 8;
        int t = tokLDS[row];
        float4* dst = (float4*)&xs[row][lane * 4];
        if (t >= 0) {
            *dst = *(const float4*)(xin + (size_t)t * DEMB + lane * 4);
        } else {
            float4 z = {0.f, 0.f, 0.f, 0.f};
            *dst = z;
        }
    }
    __syncthreads();

    const int fq = w & 3;               // f-quarter: columns fq*32 .. fq*32+31
    const int mp = w >> 2;              // m-pair: token rows mp*32 .. mp*32+31
    const int m  = lane & 15;           // A row within subtile / B,N column index
    const int hi = lane >> 4;           // lane-group: K pair select

    const float* Wp0 = Ws + (((size_t)layer * 4 + 0) * NEXP + e) * (DEMB * DEMB);
    const float* Wp1 = Ws + (((size_t)layer * 4 + 1) * NEXP + e) * (DEMB * DEMB);
    const float* Wp2 = Ws + (((size_t)layer * 4 + 2) * NEXP + e) * (DEMB * DEMB);
    const float* Wp3 = Ws + (((size_t)layer * 4 + 3) * NEXP + e) * (DEMB * DEMB);
    const float* Wp[4] = {Wp0, Wp1, Wp2, Wp3};

    // acc[j][nt][ms]: 4 linears x 2 n-tiles x 2 m-subtiles = 128 acc VGPRs
    v8f acc[4][2][2];
    #pragma unroll
    for (int j = 0; j < 4; ++j)
        #pragma unroll
        for (int nt = 0; nt < 2; ++nt)
            #pragma unroll
            for (int ms = 0; ms < 2; ++ms)
                acc[j][nt][ms] = (v8f){0.f,0.f,0.f,0.f,0.f,0.f,0.f,0.f};

    const int row0 = (mp * 2 + 0) * 16 + m;
    const int row1 = (mp * 2 + 1) * 16 + m;

    // K = 128 in 32 steps of V_WMMA_F32_16X16X4_F32.
    // Each B fragment (2 dword loads from L2-resident weights) feeds TWO wmmas.
    #pragma unroll 2
    for (int ks = 0; ks < 32; ++ks) {
        const int kb = ks * 4 + hi * 2;
        v2f a0, a1;
        a0.x = xs[row0][kb];            // contiguous pair -> ds_load_b64
        a0.y = xs[row0][kb + 1];
        a1.x = xs[row1][kb];
        a1.y = xs[row1][kb + 1];
        #pragma unroll
        for (int j = 0; j < 4; ++j) {
            const float* Wj = Wp[j];
            #pragma unroll
            for (int nt = 0; nt < 2; ++nt) {
                const int n = fq * 32 + nt * 16 + m;
                v2f bf;
                bf.x = Wj[(size_t)kb * DEMB + n];
                bf.y = Wj[(size_t)(kb + 1) * DEMB + n];
                acc[j][nt][0] = __builtin_amdgcn_wmma_f32_16x16x4_f32(
                    false, a0, false, bf, (short)0, acc[j][nt][0], false, false);
                acc[j][nt][1] = __builtin_amdgcn_wmma_f32_16x16x4_f32(
                    false, a1, false, bf, (short)0, acc[j][nt][1], false, false);
            }
        }
    }

    // In-register bias + gating + scattered store.
    // C layout: VGPR row r -> token row = ms*16 + r + 8*hi, f = fq*32 + nt*16 + (lane&15)
    const float* bp0 = bs + (((size_t)layer * 4 + 0) * NEXP + e) * DEMB;
    const float* bp1 = bs + (((size_t)layer * 4 + 1) * NEXP + e) * DEMB;
    const float* bp2 = bs + (((size_t)layer * 4 + 2) * NEXP + e) * DEMB;
    const float* bp3 = bs + (((size_t)layer * 4 + 3) * NEXP + e) * DEMB;

    #pragma unroll
    for (int nt = 0; nt < 2; ++nt) {
        const int n = fq * 32 + nt * 16 + m;
        const float bv0 = bp0[n];
        const float bv1 = bp1[n];
        const float bv2 = bp2[n];
        const float bv3 = bp3[n];
        #pragma unroll
        for (int ms = 0; ms < 2; ++ms) {
            const int rowbase = (mp * 2 + ms) * 16 + 8 * hi;
            #pragma unroll
            for (int r = 0; r < 8; ++r) {
                const int row = rowbase + r;
                const int t = tokLDS[row];
                if (t >= 0) {
                    const float sf = acc[0][nt][ms][r] + bv0;
                    const float lg = acc[1][nt][ms][r] + bv1;
                    const float lf = acc[2][nt][ms][r] + bv2;
                    const float of = acc[3][nt][ms][r] + bv3;
                    const float xv = xs[row][n];
                    const float nr = xv * sigm(sf) + tanhf(lg) * sigm(lf);
                    xout[(size_t)t * DEMB + n] = tanhf(nr) * sigm(of);
                }
            }
        }
    }
}

// ---------------- launcher ----------------

extern "C" void kernel_launch(void* const* d_in, const int* in_sizes, int n_in,
                              void* d_out, int out_size, void* d_ws, size_t ws_size,
                              hipStream_t stream) {
    const int*   positions = (const int*)d_in[0];    // [T,B] (int32 on device)
    const float* outputs   = (const float*)d_in[1];  // [T,B,128] fp32
    const float* Ws        = (const float*)d_in[2];  // [2,4,9,128,128] fp32
    const float* bs        = (const float*)d_in[3];  // [2,4,9,128] fp32
    float*       out       = (float*)d_out;          // [T,B,128] fp32

    // workspace layout (ints)
    int* counts  = (int*)d_ws;        // 9 (padded to 16)
    int* offsets = counts + 16;       // 9
    int* fill    = counts + 32;       // 9
    int* tok     = counts + 64;       // 65536 token ids, expert-segment order

    mpt_init_kernel<<<1, 32, 0, stream>>>(counts, fill);
    mpt_count_kernel<<<NTOK / 256, 256, 0, stream>>>(positions, counts);
    mpt_scan_kernel<<<1, 32, 0, stream>>>(counts, offsets);
    mpt_scatter_kernel<<<NTOK / 256, 256, 0, stream>>>(positions, offsets, fill, tok);

    dim3 grid(MAXTILES, NEXP);
    // layer 0: outputs -> d_out ; layer 1: d_out -> d_out (in-place, blocks own their rows)
    mpt_layer_kernel<<<grid, 256, 0, stream>>>(outputs, out, Ws, bs,
                                               counts, offsets, tok, 0);
    mpt_layer_kernel<<<grid, 256, 0, stream>>>(out, out, Ws, bs,
                                               counts, offsets, tok, 1);
}